// GNN_69853348102417
// MI455X (gfx1250) — compile-verified
//
#include <hip/hip_runtime.h>

typedef float v2f __attribute__((ext_vector_type(2)));
typedef float v8f __attribute__((ext_vector_type(8)));

__device__ __forceinline__ float lrelu(float v) { return v > 0.0f ? v : 0.01f * v; }

// ---------------------------------------------------------------------------
// grid-stride zero fill
// ---------------------------------------------------------------------------
__global__ void zero_kernel(float* __restrict__ p, long n) {
    long i = (long)blockIdx.x * blockDim.x + threadIdx.x;
    long stride = (long)gridDim.x * blockDim.x;
    for (; i < n; i += stride) p[i] = 0.0f;
}

// ---------------------------------------------------------------------------
// WMMA f32 GEMM:  Out[nrows,32] = X[nrows,KDIM] @ W[KDIM,32]
// One wave computes a 16x32 tile via two V_WMMA_F32_16X16X4_F32 accumulators.
// EPI: Out = lrelu(acc + Sum[row,:]/max(Cnt[row],1) + Bias)
//
// A-frag (16x4 f32, ISA 7.12.2): lanes 0-15 hold M=lane,K={0,1}; lanes 16-31
// hold M=lane-16,K={2,3}  -> contiguous float2 per lane.
// B-frag (4x16): transpose-symmetric: lane n holds K={0,1} (lanes 0-15) or
// K={2,3} (lanes 16-31) of column n.
// C/D (16x16 f32): VGPR r = row r (lanes 0-15) / row r+8 (lanes 16-31).
// ---------------------------------------------------------------------------
template <int KDIM, bool EPI>
__global__ void gemm_n32_wmma(const float* __restrict__ X,
                              const float* __restrict__ W,
                              const float* __restrict__ Sum,
                              const float* __restrict__ Cnt,
                              const float* __restrict__ Bias,
                              float* __restrict__ Out,
                              int nrows) {
    const int wave = threadIdx.x >> 5;
    const int lane = threadIdx.x & 31;
    const int half = lane >> 4;   // 0: lanes 0-15, 1: lanes 16-31
    const int mn   = lane & 15;   // A-row / B-col within the 16-wide tile
    const int tile = blockIdx.x * (blockDim.x >> 5) + wave;
    const int row0 = tile * 16;
    if (row0 >= nrows) return;    // uniform per wave; EXEC stays all-1s for WMMA

    v8f c0 = {};  // output columns 0..15
    v8f c1 = {};  // output columns 16..31
    const float* xrow = X + (size_t)(row0 + mn) * KDIM;

#pragma unroll
    for (int k = 0; k < KDIM; k += 4) {
        v2f a;
        a.x = xrow[k + 2 * half];
        a.y = xrow[k + 2 * half + 1];

        const float* wk = W + (size_t)(k + 2 * half) * 32;
        v2f b0, b1;
        b0.x = wk[mn];            // W[k+2h  ][n]
        b0.y = wk[32 + mn];       // W[k+2h+1][n]
        b1.x = wk[mn + 16];
        b1.y = wk[32 + mn + 16];

        c0 = __builtin_amdgcn_wmma_f32_16x16x4_f32(false, a, false, b0,
                                                   (short)0, c0, false, false);
        c1 = __builtin_amdgcn_wmma_f32_16x16x4_f32(false, a, false, b1,
                                                   (short)0, c1, false, false);
    }

    float bias0 = 0.0f, bias1 = 0.0f;
    if (EPI) { bias0 = Bias[mn]; bias1 = Bias[mn + 16]; }

#pragma unroll
    for (int r = 0; r < 8; ++r) {
        const int row = row0 + r + 8 * half;
        float v0 = c0[r];
        float v1 = c1[r];
        if (EPI) {
            const float inv = 1.0f / fmaxf(Cnt[row], 1.0f);
            v0 = lrelu(v0 + Sum[(size_t)row * 32 + mn]      * inv + bias0);
            v1 = lrelu(v1 + Sum[(size_t)row * 32 + mn + 16] * inv + bias1);
        }
        Out[(size_t)row * 32 + mn]      = v0;
        Out[(size_t)row * 32 + mn + 16] = v1;
    }
}

// ---------------------------------------------------------------------------
// Edge scatter (segment mean numerator + count). One wave per edge, one lane
// per feature; src/dst loads are wave-uniform, payload loads coalesced 128B.
// ---------------------------------------------------------------------------
__global__ void scatter_single(const int* __restrict__ src, const int* __restrict__ dst,
                               const float* __restrict__ Y,
                               float* __restrict__ Sum, float* __restrict__ Cnt,
                               int nedges) {
    const long idx = (long)blockIdx.x * blockDim.x + threadIdx.x;
    const long e = idx >> 5;
    const int  f = (int)(idx & 31);
    if (e >= nedges) return;
    const int s = src[e];
    const int d = dst[e];
    atomicAdd(&Sum[(size_t)d * 32 + f], Y[(size_t)s * 32 + f]);
    if (f == 0) atomicAdd(&Cnt[d], 1.0f);
}

// fused: both p2c aggregations share one edge list
__global__ void scatter_dual(const int* __restrict__ src, const int* __restrict__ dst,
                             const float* __restrict__ YA, const float* __restrict__ YB,
                             float* __restrict__ SumA, float* __restrict__ SumB,
                             float* __restrict__ Cnt, int nedges) {
    const long idx = (long)blockIdx.x * blockDim.x + threadIdx.x;
    const long e = idx >> 5;
    const int  f = (int)(idx & 31);
    if (e >= nedges) return;
    const int s = src[e];
    const int d = dst[e];
    atomicAdd(&SumA[(size_t)d * 32 + f], YA[(size_t)s * 32 + f]);
    atomicAdd(&SumB[(size_t)d * 32 + f], YB[(size_t)s * 32 + f]);
    if (f == 0) atomicAdd(&Cnt[d], 1.0f);
}

// ---------------------------------------------------------------------------
// Final head: Out[n,2] = G[n,32] @ Wl[32,2] + Bl   (trivially small)
// ---------------------------------------------------------------------------
__global__ void head_kernel(const float* __restrict__ G, const float* __restrict__ Wl,
                            const float* __restrict__ Bl, float* __restrict__ Out, int n) {
    const int i = blockIdx.x * blockDim.x + threadIdx.x;
    if (i >= n) return;
    float a0 = Bl[0], a1 = Bl[1];
    const float* g = G + (size_t)i * 32;
#pragma unroll
    for (int h = 0; h < 32; ++h) {
        const float gv = g[h];
        a0 += gv * Wl[h * 2 + 0];
        a1 += gv * Wl[h * 2 + 1];
    }
    Out[i * 2 + 0] = a0;
    Out[i * 2 + 1] = a1;
}

// ---------------------------------------------------------------------------
static inline int cdiv(long a, long b) { return (int)((a + b - 1) / b); }

extern "C" void kernel_launch(void* const* d_in, const int* in_sizes, int n_in,
                              void* d_out, int out_size, void* d_ws, size_t ws_size,
                              hipStream_t stream) {
    (void)n_in; (void)out_size; (void)ws_size;

    const float* x_paper  = (const float*)d_in[0];
    const float* x_author = (const float*)d_in[1];
    const float* x_conf   = (const float*)d_in[2];
    const int*   src_a2p  = (const int*)d_in[5];
    const int*   dst_a2p  = (const int*)d_in[6];
    const int*   src_p2c  = (const int*)d_in[7];
    const int*   dst_p2c  = (const int*)d_in[8];
    const float* w1l_a2p  = (const float*)d_in[12];
    const float* w1r_a2p  = (const float*)d_in[13];
    const float* b1_a2p   = (const float*)d_in[14];
    const float* w1l_p2c  = (const float*)d_in[15];
    const float* w1r_p2c  = (const float*)d_in[16];
    const float* b1_p2c   = (const float*)d_in[17];
    const float* w2l_p2c  = (const float*)d_in[24];
    const float* w2r_p2c  = (const float*)d_in[25];
    const float* b2_p2c   = (const float*)d_in[26];
    const float* lin_w    = (const float*)d_in[27];
    const float* lin_b    = (const float*)d_in[28];

    const int NP = in_sizes[0] / 64;
    const int NA = in_sizes[1] / 64;
    const int NC = in_sizes[2] / 64;
    const int E_A2P = in_sizes[5];
    const int E_P2C = in_sizes[7];

    // workspace layout (floats); bufY and sum_p are reused across phases
    float* ws = (float*)d_ws;
    size_t off = 0;
    const size_t nmax = (size_t)(NP > NA ? NP : NA);
    float* bufY    = ws + off; off += nmax * 32;        // y_author -> y_pc1
    float* sum_p   = ws + off; off += (size_t)NP * 32;  // agg(paper) -> z_pc2
    float* cnt_p   = ws + off; off += (size_t)NP;
    float* h_paper = ws + off; off += (size_t)NP * 32;
    float* sumA    = ws + off; off += (size_t)NC * 32;
    float* sumB    = ws + off; off += (size_t)NC * 32;
    float* cnt_c   = ws + off; off += (size_t)NC;
    float* h_conf  = ws + off; off += (size_t)NC * 32;
    float* g_conf  = ws + off; off += (size_t)NC * 32;

    const dim3 blk(256);
    const int ZG = 2048;  // grid-stride zero fill

    // --- layer 1, a2p relation: h_paper ---
    zero_kernel<<<ZG, blk, 0, stream>>>(sum_p, (long)NP * 32);
    zero_kernel<<<ZG, blk, 0, stream>>>(cnt_p, (long)NP);

    // y_author = x_author @ w1l_a2p
    gemm_n32_wmma<64, false><<<cdiv((long)NA / 16, 8), blk, 0, stream>>>(
        x_author, w1l_a2p, nullptr, nullptr, nullptr, bufY, NA);

    scatter_single<<<cdiv((long)E_A2P * 32, 256), blk, 0, stream>>>(
        src_a2p, dst_a2p, bufY, sum_p, cnt_p, E_A2P);

    // h_paper = lrelu(agg + x_paper @ w1r_a2p + b1)
    gemm_n32_wmma<64, true><<<cdiv((long)NP / 16, 8), blk, 0, stream>>>(
        x_paper, w1r_a2p, sum_p, cnt_p, b1_a2p, h_paper, NP);

    // --- p2c pre-transforms (bufY and sum_p are now free to reuse) ---
    gemm_n32_wmma<64, false><<<cdiv((long)NP / 16, 8), blk, 0, stream>>>(
        x_paper, w1l_p2c, nullptr, nullptr, nullptr, bufY, NP);       // y_pc1
    gemm_n32_wmma<32, false><<<cdiv((long)NP / 16, 8), blk, 0, stream>>>(
        h_paper, w2l_p2c, nullptr, nullptr, nullptr, sum_p, NP);      // z_pc2

    // --- fused p2c aggregation for both layers ---
    zero_kernel<<<ZG, blk, 0, stream>>>(sumA, (long)NC * 32);
    zero_kernel<<<ZG, blk, 0, stream>>>(sumB, (long)NC * 32);
    zero_kernel<<<ZG, blk, 0, stream>>>(cnt_c, (long)NC);

    scatter_dual<<<cdiv((long)E_P2C * 32, 256), blk, 0, stream>>>(
        src_p2c, dst_p2c, bufY, sum_p, sumA, sumB, cnt_c, E_P2C);

    // h_conf = lrelu(aggA + x_conf @ w1r_p2c + b1)
    gemm_n32_wmma<64, true><<<cdiv((long)NC / 16, 8), blk, 0, stream>>>(
        x_conf, w1r_p2c, sumA, cnt_c, b1_p2c, h_conf, NC);
    // g_conf = lrelu(aggB + h_conf @ w2r_p2c + b2)
    gemm_n32_wmma<32, true><<<cdiv((long)NC / 16, 8), blk, 0, stream>>>(
        h_conf, w2r_p2c, sumB, cnt_c, b2_p2c, g_conf, NC);

    // out = g_conf @ lin_w + lin_b
    head_kernel<<<cdiv(NC, 256), blk, 0, stream>>>(
        g_conf, lin_w, lin_b, (float*)d_out, NC);
}